// DSN_8985071583873
// MI455X (gfx1250) — compile-verified
//
#include <hip/hip_runtime.h>
#include <math.h>

#define B_ 2
#define N_ 20000
#define M_ 1024
#define C_ 256
#define BN_EPS 1e-5f

typedef __attribute__((ext_vector_type(2))) float v2f;
typedef __attribute__((ext_vector_type(4))) float v4f;
typedef __attribute__((ext_vector_type(8))) float v8f;

// ---------------------------------------------------------------------------
// Kernel 1: x[b,o,m] = relu( BN( W1 @ feat + b1 ) )   (one head)
// V_WMMA_F32_16X16X4_F32, exact fp32. Each wave owns a 16x64 output strip
// (4 accumulators) so the A fragment is reused 4x. K is walked in steps of 8
// with a permuted K-slot mapping: hardware slots {hi*2+vgpr} are assigned
// logical k = k0 + 4*hi + {0,1} (step 0) and k0 + 4*hi + {2,3} (step 1), so
// each lane's A data for BOTH steps is one contiguous float4 (b128 load).
// A and B use the same slot mapping, so the sum over K is unchanged.
// grid = (M/512, C/16, B), block = 256 (8 waves).
// ---------------------------------------------------------------------------
__global__ __launch_bounds__(256) void head_gemm_bn_relu(
    const float* __restrict__ W1,    // C x C  (row = out channel, col = in channel)
    const float* __restrict__ feat,  // B x C x M
    const float* __restrict__ b1,
    const float* __restrict__ gamma,
    const float* __restrict__ beta,
    const float* __restrict__ mu,
    const float* __restrict__ var,
    float* __restrict__ x)           // B x C x M
{
    const int lane = threadIdx.x & 31;
    const int wave = threadIdx.x >> 5;
    const int b      = blockIdx.z;
    const int o_base = blockIdx.y * 16;
    const int m_base = (blockIdx.x * 8 + wave) * 64;   // 4 tiles of 16

    const int r  = lane & 15;   // row-in-tile (A) / col-in-tile (B,D)
    const int hi = lane >> 4;   // K-slot group select

    const float* featB = feat + (size_t)b * C_ * M_;
    const float* arow  = W1 + (size_t)(o_base + r) * C_;

    v8f acc[4] = {};
    for (int k = 0; k < C_; k += 8) {
        // A: W1[o_base+r, k + 4*hi .. k + 4*hi + 3]  (one b128, feeds 2 steps)
        const v4f a4 = *(const v4f*)(arow + k + 4 * hi);
        const v2f a_lo = {a4.x, a4.y};   // logical k slots {k+4hi,   k+4hi+1}
        const v2f a_hi = {a4.z, a4.w};   // logical k slots {k+4hi+2, k+4hi+3}

        const float* bp = featB + (size_t)(k + 4 * hi) * M_ + (m_base + r);
#pragma unroll
        for (int t = 0; t < 4; ++t) {
            const int mo = 16 * t;
            v2f b01, b23;
            b01.x = bp[mo];
            b01.y = bp[(size_t)M_ + mo];
            b23.x = bp[(size_t)2 * M_ + mo];
            b23.y = bp[(size_t)3 * M_ + mo];
            acc[t] = __builtin_amdgcn_wmma_f32_16x16x4_f32(
                false, a_lo, false, b01, (short)0, acc[t], false, false);
            acc[t] = __builtin_amdgcn_wmma_f32_16x16x4_f32(
                false, a_hi, false, b23, (short)0, acc[t], false, false);
        }
    }

    // Epilogue: fold bias + batchnorm into one affine per output channel.
    float scl[8], off[8];
#pragma unroll
    for (int v = 0; v < 8; ++v) {
        const int o = o_base + v + 8 * hi;           // D layout: row = v + 8*(lane>>4)
        const float s = gamma[o] * rsqrtf(var[o] + BN_EPS);
        scl[v] = s;
        off[v] = s * (b1[o] - mu[o]) + beta[o];
    }
    float* xB = x + (size_t)b * C_ * M_;
#pragma unroll
    for (int t = 0; t < 4; ++t) {
        const int m = m_base + 16 * t + r;
#pragma unroll
        for (int v = 0; v < 8; ++v) {
            const int o = o_base + v + 8 * hi;
            const float val = fmaxf(fmaf(scl[v], acc[t][v], off[v]), 0.0f);
            xB[(size_t)o * M_ + m] = val;
        }
    }
}

// ---------------------------------------------------------------------------
// Kernel 2: so5[b, 0:2, m] = w2f @ xf + b2f ; so5[b, 2:5, m] = w2c @ xc + b2c
// One thread per (b, m); 5 running dot products over C=256.
// ---------------------------------------------------------------------------
__global__ __launch_bounds__(256) void seed_head_out(
    const float* __restrict__ xf, const float* __restrict__ xc,
    const float* __restrict__ w2f, const float* __restrict__ b2f,
    const float* __restrict__ w2c, const float* __restrict__ b2c,
    float* __restrict__ so5)        // B x 5 x M
{
    const int t = blockIdx.x * blockDim.x + threadIdx.x;  // 0 .. B*M-1
    const int b = t / M_;
    const int m = t % M_;
    const float* xfB = xf + (size_t)b * C_ * M_ + m;
    const float* xcB = xc + (size_t)b * C_ * M_ + m;
    float a0 = b2f[0], a1 = b2f[1];
    float a2 = b2c[0], a3 = b2c[1], a4 = b2c[2];
    for (int c = 0; c < C_; ++c) {
        const float vf = xfB[(size_t)c * M_];
        const float vc = xcB[(size_t)c * M_];
        a0 = fmaf(w2f[c],           vf, a0);
        a1 = fmaf(w2f[C_ + c],      vf, a1);
        a2 = fmaf(w2c[c],           vc, a2);
        a3 = fmaf(w2c[C_ + c],      vc, a3);
        a4 = fmaf(w2c[2 * C_ + c],  vc, a4);
    }
    float* o = so5 + (size_t)b * 5 * M_ + m;
    o[0 * M_] = a0;
    o[1 * M_] = a1;
    o[2 * M_] = a2;
    o[3 * M_] = a3;
    o[4 * M_] = a4;
}

// ---------------------------------------------------------------------------
// Kernel 3: fused 3-NN + inverse-distance interpolation.
// Seeds staged in LDS (12 KB); one thread per query point keeps top-3 in
// registers, then gathers the 5 channels from the L2-resident so5 table.
// ---------------------------------------------------------------------------
#define PTS_PER_BLOCK 256
__global__ __launch_bounds__(PTS_PER_BLOCK) void knn_interp(
    const float* __restrict__ pts,    // B x N x 3
    const float* __restrict__ seeds,  // B x M x 3
    const float* __restrict__ so5,    // B x 5 x M
    float* __restrict__ out)          // B x 5 x N
{
    __shared__ float sx[M_], sy[M_], sz[M_];
    const int blocksPerBatch = (N_ + PTS_PER_BLOCK - 1) / PTS_PER_BLOCK;
    const int b = blockIdx.x / blocksPerBatch;
    const int n = (blockIdx.x % blocksPerBatch) * PTS_PER_BLOCK + threadIdx.x;

    const float* sB = seeds + (size_t)b * M_ * 3;
    for (int i = threadIdx.x; i < M_; i += PTS_PER_BLOCK) {
        sx[i] = sB[i * 3 + 0];
        sy[i] = sB[i * 3 + 1];
        sz[i] = sB[i * 3 + 2];
    }
    __syncthreads();
    if (n >= N_) return;

    const float* p = pts + ((size_t)b * N_ + n) * 3;
    const float px = p[0], py = p[1], pz = p[2];

    float d0 = INFINITY, d1 = INFINITY, d2 = INFINITY;
    int   i0 = 0,        i1 = 0,        i2 = 0;
    for (int m = 0; m < M_; ++m) {
        const float dx = sx[m] - px;
        const float dy = sy[m] - py;
        const float dz = sz[m] - pz;
        const float d = fmaf(dx, dx, fmaf(dy, dy, dz * dz));
        if (d < d2) {
            if (d < d1) {
                d2 = d1; i2 = i1;
                if (d < d0) { d1 = d0; i1 = i0; d0 = d; i0 = m; }
                else        { d1 = d;  i1 = m; }
            } else { d2 = d; i2 = m; }
        }
    }

    const float r0 = 1.0f / (sqrtf(fmaxf(d0, 1e-12f)) + 1e-8f);
    const float r1 = 1.0f / (sqrtf(fmaxf(d1, 1e-12f)) + 1e-8f);
    const float r2 = 1.0f / (sqrtf(fmaxf(d2, 1e-12f)) + 1e-8f);
    const float rs = 1.0f / (r0 + r1 + r2);
    const float w0 = r0 * rs, w1 = r1 * rs, w2 = r2 * rs;

    const float* soB = so5 + (size_t)b * 5 * M_;
    float* oB = out + (size_t)b * 5 * N_ + n;
#pragma unroll
    for (int o = 0; o < 5; ++o) {
        const float* row = soB + (size_t)o * M_;
        oB[(size_t)o * N_] = w0 * row[i0] + w1 * row[i1] + w2 * row[i2];
    }
}

// ---------------------------------------------------------------------------
extern "C" void kernel_launch(void* const* d_in, const int* in_sizes, int n_in,
                              void* d_out, int out_size, void* d_ws, size_t ws_size,
                              hipStream_t stream) {
    const float* pts      = (const float*)d_in[0];   // (B,N,3)
    const float* seed_xyz = (const float*)d_in[1];   // (B,M,3)
    const float* feat     = (const float*)d_in[2];   // (B,C,M)
    const float* w1f  = (const float*)d_in[3];
    const float* b1f  = (const float*)d_in[4];
    const float* g1f  = (const float*)d_in[5];
    const float* be1f = (const float*)d_in[6];
    const float* mu1f = (const float*)d_in[7];
    const float* v1f  = (const float*)d_in[8];
    const float* w2f  = (const float*)d_in[9];
    const float* b2f  = (const float*)d_in[10];
    const float* w1c  = (const float*)d_in[11];
    const float* b1c  = (const float*)d_in[12];
    const float* g1c  = (const float*)d_in[13];
    const float* be1c = (const float*)d_in[14];
    const float* mu1c = (const float*)d_in[15];
    const float* v1c  = (const float*)d_in[16];
    const float* w2c  = (const float*)d_in[17];
    const float* b2c  = (const float*)d_in[18];

    float* ws  = (float*)d_ws;
    float* xf  = ws;                                   // B*C*M
    float* xc  = ws + (size_t)B_ * C_ * M_;            // B*C*M
    float* so5 = ws + 2 * (size_t)B_ * C_ * M_;        // B*5*M

    dim3 gGemm(M_ / 512, C_ / 16, B_);                 // 8 waves/block, 16x64 strip/wave
    head_gemm_bn_relu<<<gGemm, 256, 0, stream>>>(w1f, feat, b1f, g1f, be1f, mu1f, v1f, xf);
    head_gemm_bn_relu<<<gGemm, 256, 0, stream>>>(w1c, feat, b1c, g1c, be1c, mu1c, v1c, xc);

    seed_head_out<<<(B_ * M_) / 256, 256, 0, stream>>>(xf, xc, w2f, b2f, w2c, b2c, so5);

    const int blocksPerBatch = (N_ + PTS_PER_BLOCK - 1) / PTS_PER_BLOCK;
    knn_interp<<<B_ * blocksPerBatch, PTS_PER_BLOCK, 0, stream>>>(pts, seed_xyz, so5,
                                                                  (float*)d_out);
}